// IrrepsLinear_19688130084902
// MI455X (gfx1250) — compile-verified
//
#include <hip/hip_runtime.h>

typedef __attribute__((ext_vector_type(2))) float v2f;
typedef __attribute__((ext_vector_type(4))) float v4f;
typedef __attribute__((ext_vector_type(8))) float v8f;

#define MUL0 128
#define MUL1 64
#define MUL2 32
#define D0   128
#define D1   192
#define D2   160
#define DT   480
#define LSI  484   // LDS row stride (floats), input slab: 16B aligned, conflict-free
#define LSO  484   // LDS row stride (floats), output slab

__device__ __forceinline__ v8f wmma_f32(v2f a, v2f b, v8f c) {
  // D = A(16x4 f32) * B(4x16 f32) + C(16x16 f32)
  return __builtin_amdgcn_wmma_f32_16x16x4_f32(
      /*neg_a=*/false, a, /*neg_b=*/false, b,
      /*c_mod=*/(short)0, c, /*reuse_a=*/false, /*reuse_b=*/false);
}

__global__ __launch_bounds__(256)
void irreps_linear_kernel(const float* __restrict__ x,
                          const float* __restrict__ modal_attr,
                          const float* __restrict__ W0,
                          const float* __restrict__ W0m,
                          const float* __restrict__ W1,
                          const float* __restrict__ W2,
                          const int*   __restrict__ batch,
                          float*       __restrict__ out,
                          int Nrows, int nslabs)
{
  __shared__ __align__(16) float lin [16 * LSI];
  __shared__ __align__(16) float lout[16 * LSO];
  __shared__ float smod[32];   // [row*2 + {0,1}] modal values per slab row

  const int tid  = threadIdx.x;
  const int wave = tid >> 5;
  const int lane = tid & 31;
  const int l16  = lane & 15;
  const int half = lane >> 4;

  // ---- static tile assignment (30 16-col output tiles over 8 waves) ----
  // out0: tile nt = wave            (8 tiles, K=128)
  // out1: tiles j = wave, 8+wave(<4 only); j -> m=j/4, nt=j%4   (12 tiles, K=64)
  // out2: tiles j = wave, 8+wave(<2 only); j -> m=j/2, nt=j%2   (10 tiles, K=32)
  const int  col0  = wave * 16 + l16;
  const int  m1    = wave >> 2;
  const int  col1  = (wave & 3) * 16 + l16;
  const int  oc1   = D0 + 3 * col1 + m1;
  const bool has1b = (wave < 4);
  const int  col1b = wave * 16 + l16;          // m=2, nt=wave
  const int  oc1b  = D0 + 3 * col1b + 2;
  const int  m2    = wave >> 1;
  const int  col2  = (wave & 1) * 16 + l16;
  const int  oc2   = D0 + D1 + 5 * col2 + m2;
  const bool has2b = (wave < 2);
  const int  col2b = wave * 16 + l16;          // m=4, nt=wave
  const int  oc2b  = D0 + D1 + 5 * col2b + 4;

  // ---- preload all B fragments into registers (weights are tiny & reused
  //      across thousands of row-slabs). B frag layout for 16x16x4 f32:
  //      vgpr0 = W[k0+2*half][col], vgpr1 = W[k0+1+2*half][col]. ----
  v2f b0[32];
  #pragma unroll
  for (int ks = 0; ks < 32; ++ks) {
    int k = 4 * ks + 2 * half;
    b0[ks].x = W0[k * MUL0 + col0];
    b0[ks].y = W0[(k + 1) * MUL0 + col0];
  }
  v2f b1a[16], b1b[16];
  #pragma unroll
  for (int ks = 0; ks < 16; ++ks) {
    int k = 4 * ks + 2 * half;
    b1a[ks].x = W1[k * MUL1 + col1];
    b1a[ks].y = W1[(k + 1) * MUL1 + col1];
    b1b[ks].x = has1b ? W1[k * MUL1 + col1b]       : 0.0f;
    b1b[ks].y = has1b ? W1[(k + 1) * MUL1 + col1b] : 0.0f;
  }
  v2f b2a[8], b2b[8];
  #pragma unroll
  for (int ks = 0; ks < 8; ++ks) {
    int k = 4 * ks + 2 * half;
    b2a[ks].x = W2[k * MUL2 + col2];
    b2a[ks].y = W2[(k + 1) * MUL2 + col2];
    b2b[ks].x = has2b ? W2[k * MUL2 + col2b]       : 0.0f;
    b2b[ks].y = has2b ? W2[(k + 1) * MUL2 + col2b] : 0.0f;
  }
  const float wm0 = W0m[col0];
  const float wm1 = W0m[MUL0 + col0];

  const float INV0 = 0.08770580f;   // 1/sqrt(130)
  const float INV1 = 0.125f;        // 1/sqrt(64)
  const float INV2 = 0.17677670f;   // 1/sqrt(32)

  for (int s = blockIdx.x; s < nslabs; s += gridDim.x) {
    const int base_row = s * 16;

    // ---- stage 16x480 x-slab into LDS (coalesced float4) ----
    for (int idx = tid; idx < 16 * 120; idx += 256) {
      int r = idx / 120, c4 = idx % 120;
      v4f v = {0.0f, 0.0f, 0.0f, 0.0f};
      int row = base_row + r;
      if (row < Nrows) v = *(const v4f*)(x + (size_t)row * DT + c4 * 4);
      *(v4f*)(lin + r * LSI + c4 * 4) = v;
    }
    if (tid < 16) {
      int row = base_row + tid;
      float a0 = 0.0f, a1 = 0.0f;
      if (row < Nrows) {
        int b = batch[row];
        a0 = modal_attr[2 * b];
        a1 = modal_attr[2 * b + 1];
      }
      smod[2 * tid] = a0;
      smod[2 * tid + 1] = a1;
    }
    // ---- prefetch next grid-stride slab toward L2/WGP$ (global_prefetch_b8):
    //      one 128B line per thread covers the whole 30KB next slab ----
    {
      int pre = s + gridDim.x;
      if (pre < nslabs) {
        size_t pbase = (size_t)pre * 16 * DT;
        size_t poff  = (size_t)tid * 32;            // 32 floats = 128B per thread
        if (poff < (size_t)16 * DT)
          __builtin_prefetch(x + pbase + poff, 0, 3);
      }
    }
    __syncthreads();

    // ---- out0 tile: C initialized with modal @ W0m, then 32 K-steps ----
    {
      v8f c;
      #pragma unroll
      for (int r = 0; r < 8; ++r) {
        int M = r + 8 * half;                 // C layout: vgpr r, lane half
        c[r] = smod[2 * M] * wm0 + smod[2 * M + 1] * wm1;
      }
      #pragma unroll
      for (int ks = 0; ks < 32; ++ks) {
        v2f a = *(const v2f*)(lin + l16 * LSI + 4 * ks + 2 * half);
        c = wmma_f32(a, b0[ks], c);
      }
      #pragma unroll
      for (int r = 0; r < 8; ++r)
        lout[(r + 8 * half) * LSO + col0] = c[r] * INV0;
    }

    // ---- out1 tiles (x1 elements at col D0 + 3*i + m) ----
    {
      v8f c = {};
      #pragma unroll
      for (int ks = 0; ks < 16; ++ks) {
        int i = 4 * ks + 2 * half;
        const float* p = lin + l16 * LSI + D0 + 3 * i + m1;
        v2f a; a.x = p[0]; a.y = p[3];
        c = wmma_f32(a, b1a[ks], c);
      }
      #pragma unroll
      for (int r = 0; r < 8; ++r)
        lout[(r + 8 * half) * LSO + oc1] = c[r] * INV1;
    }
    if (has1b) {
      v8f c = {};
      #pragma unroll
      for (int ks = 0; ks < 16; ++ks) {
        int i = 4 * ks + 2 * half;
        const float* p = lin + l16 * LSI + D0 + 3 * i + 2;
        v2f a; a.x = p[0]; a.y = p[3];
        c = wmma_f32(a, b1b[ks], c);
      }
      #pragma unroll
      for (int r = 0; r < 8; ++r)
        lout[(r + 8 * half) * LSO + oc1b] = c[r] * INV1;
    }

    // ---- out2 tiles (x2 elements at col D0+D1 + 5*i + m) ----
    {
      v8f c = {};
      #pragma unroll
      for (int ks = 0; ks < 8; ++ks) {
        int i = 4 * ks + 2 * half;
        const float* p = lin + l16 * LSI + D0 + D1 + 5 * i + m2;
        v2f a; a.x = p[0]; a.y = p[5];
        c = wmma_f32(a, b2a[ks], c);
      }
      #pragma unroll
      for (int r = 0; r < 8; ++r)
        lout[(r + 8 * half) * LSO + oc2] = c[r] * INV2;
    }
    if (has2b) {
      v8f c = {};
      #pragma unroll
      for (int ks = 0; ks < 8; ++ks) {
        int i = 4 * ks + 2 * half;
        const float* p = lin + l16 * LSI + D0 + D1 + 5 * i + 4;
        v2f a; a.x = p[0]; a.y = p[5];
        c = wmma_f32(a, b2b[ks], c);
      }
      #pragma unroll
      for (int r = 0; r < 8; ++r)
        lout[(r + 8 * half) * LSO + oc2b] = c[r] * INV2;
    }
    __syncthreads();

    // ---- coalesced float4 store of the 16x480 output slab ----
    for (int idx = tid; idx < 16 * 120; idx += 256) {
      int r = idx / 120, c4 = idx % 120;
      int row = base_row + r;
      if (row < Nrows) {
        v4f v = *(const v4f*)(lout + r * LSO + c4 * 4);
        *(v4f*)(out + (size_t)row * DT + c4 * 4) = v;
      }
    }
    // no trailing barrier needed: next fill writes lin (last read before the
    // mid barrier) and lout is rewritten only after the next post-fill barrier.
  }
}

extern "C" void kernel_launch(void* const* d_in, const int* in_sizes, int n_in,
                              void* d_out, int out_size, void* d_ws, size_t ws_size,
                              hipStream_t stream) {
  const float* x     = (const float*)d_in[0];
  const float* modal = (const float*)d_in[1];
  const float* W0    = (const float*)d_in[2];
  const float* W0m   = (const float*)d_in[3];
  const float* W1    = (const float*)d_in[4];
  const float* W2    = (const float*)d_in[5];
  const int*   batch = (const int*)d_in[6];
  float*       out   = (float*)d_out;

  const int Nrows  = in_sizes[0] / DT;
  const int nslabs = (Nrows + 15) / 16;
  const int blocks = nslabs < 2048 ? nslabs : 2048;

  irreps_linear_kernel<<<blocks, 256, 0, stream>>>(
      x, modal, W0, W0m, W1, W2, batch, out, Nrows, nslabs);
}